// OptModel_YW_75874892251404
// MI455X (gfx1250) — compile-verified
//
#include <hip/hip_runtime.h>

typedef float v2f __attribute__((ext_vector_type(2)));
typedef float v8f __attribute__((ext_vector_type(8)));

#define NJ    25
#define NK    15
#define DOF   78
#define NR    45      // active residual rows (env block)
#define KPAD  48      // padded K for WMMA (multiple of 4)
#define MPAD  80      // padded M/N (multiple of 16)
#define JST   80      // J row stride (J is KPAD x JST)
#define AST   80      // A row stride
#define DAMPING 0.001f
#define STEP    0.7f
#define ITERS   15

__constant__ int c_parents[NJ] = {-1,0,0,0,1,2,3,4,5,6,7,8,9,9,9,12,13,14,16,17,18,19,20,21,15};
__constant__ int c_lb[NK]      = {0,1,3,4,6,8,9,13,15,16,17,20,22,23,24};

// Rodrigues: axis-angle -> rotation matrix (scalars only; no private arrays).
__device__ __forceinline__ void rod9(float r0, float r1, float r2,
    float& R00, float& R01, float& R02,
    float& R10, float& R11, float& R12,
    float& R20, float& R21, float& R22)
{
    float t2  = r0*r0 + r1*r1 + r2*r2 + 1e-12f;
    float ang = sqrtf(t2);
    float s   = sinf(ang);
    float c   = cosf(ang);
    float inv = 1.0f / ang;
    float x = r0*inv, y = r1*inv, z = r2*inv;
    float C = 1.0f - c;
    R00 = c + x*x*C;   R01 = x*y*C - z*s; R02 = x*z*C + y*s;
    R10 = y*x*C + z*s; R11 = c + y*y*C;   R12 = y*z*C - x*s;
    R20 = z*x*C - y*s; R21 = z*y*C + x*s; R22 = c + z*z*C;
}

__device__ __forceinline__ float sgnf(float v) {
    return (v > 0.0f ? 1.0f : 0.0f) - (v < 0.0f ? 1.0f : 0.0f);
}

__global__ __launch_bounds__(32)
void lm_ik_kernel(const float* __restrict__ theta0,   // (NF, 78)
                  const float* __restrict__ tar,      // (NF, 15, 3)
                  const float* __restrict__ qres,     // (NF, 78)
                  const float* __restrict__ offs,     // (25, 3)
                  float* __restrict__ out)            // (NF, 78)
{
    __shared__ float sJ[KPAD * JST];   // env Jacobian, padded (rows 45..47, cols 78..79 stay 0)
    __shared__ float sA[MPAD * AST];   // normal matrix
    __shared__ float sTheta[MPAD];
    __shared__ float sRw[NJ * 9];      // world rotations
    __shared__ float sPw[NJ * 3];      // world positions
    __shared__ float sOff[NJ * 3];
    __shared__ float sRenv[KPAD];      // |pw - tar| residual values
    __shared__ float sSgn[KPAD];       // sign(pw - tar)
    __shared__ float sB[MPAD];         // rhs, then back-sub accumulator
    __shared__ float sF[MPAD];         // elimination factors
    __shared__ float sDelta[MPAD];

    const int lane = threadIdx.x;
    const int f    = blockIdx.x;

    for (int m = lane; m < MPAD; m += 32)
        sTheta[m] = (m < DOF) ? theta0[f * DOF + m] : 0.0f;
    for (int i = lane; i < NJ * 3; i += 32)
        sOff[i] = offs[i];
    for (int i = lane; i < KPAD * JST; i += 32)
        sJ[i] = 0.0f;
    __syncthreads();

    for (int it = 0; it < ITERS; ++it) {
        // ---------------- Forward kinematics (serial chain; lanes lockstep) ----
        {
            float R00,R01,R02,R10,R11,R12,R20,R21,R22;
            rod9(sTheta[3], sTheta[4], sTheta[5], R00,R01,R02,R10,R11,R12,R20,R21,R22);
            if (lane == 0) {
                sRw[0]=R00; sRw[1]=R01; sRw[2]=R02;
                sRw[3]=R10; sRw[4]=R11; sRw[5]=R12;
                sRw[6]=R20; sRw[7]=R21; sRw[8]=R22;
                sPw[0]=sTheta[0]; sPw[1]=sTheta[1]; sPw[2]=sTheta[2];
            }
        }
        __syncthreads();
        for (int j = 1; j < NJ; ++j) {
            int pj = c_parents[j];
            float L00,L01,L02,L10,L11,L12,L20,L21,L22;
            rod9(sTheta[3+3*j], sTheta[4+3*j], sTheta[5+3*j],
                 L00,L01,L02,L10,L11,L12,L20,L21,L22);
            float P00=sRw[pj*9+0], P01=sRw[pj*9+1], P02=sRw[pj*9+2];
            float P10=sRw[pj*9+3], P11=sRw[pj*9+4], P12=sRw[pj*9+5];
            float P20=sRw[pj*9+6], P21=sRw[pj*9+7], P22=sRw[pj*9+8];
            float W00=P00*L00+P01*L10+P02*L20, W01=P00*L01+P01*L11+P02*L21, W02=P00*L02+P01*L12+P02*L22;
            float W10=P10*L00+P11*L10+P12*L20, W11=P10*L01+P11*L11+P12*L21, W12=P10*L02+P11*L12+P12*L22;
            float W20=P20*L00+P21*L10+P22*L20, W21=P20*L01+P21*L11+P22*L21, W22=P20*L02+P21*L12+P22*L22;
            float o0=sOff[j*3+0], o1=sOff[j*3+1], o2=sOff[j*3+2];
            float px=sPw[pj*3+0]+P00*o0+P01*o1+P02*o2;
            float py=sPw[pj*3+1]+P10*o0+P11*o1+P12*o2;
            float pz=sPw[pj*3+2]+P20*o0+P21*o1+P22*o2;
            if (lane == 0) {
                sRw[j*9+0]=W00; sRw[j*9+1]=W01; sRw[j*9+2]=W02;
                sRw[j*9+3]=W10; sRw[j*9+4]=W11; sRw[j*9+5]=W12;
                sRw[j*9+6]=W20; sRw[j*9+7]=W21; sRw[j*9+8]=W22;
                sPw[j*3+0]=px; sPw[j*3+1]=py; sPw[j*3+2]=pz;
            }
            __syncthreads();
        }

        // ---------------- Environment residual + sign ------------------------
        for (int i = lane; i < KPAD; i += 32) {
            float v = 0.0f, s = 0.0f;
            if (i < NR) {
                int kk = i / 3, a = i - 3 * kk;
                int k  = c_lb[kk];
                float d = sPw[k*3 + a] - tar[f * NR + i];
                s = sgnf(d);
                v = fabsf(d);
            }
            sRenv[i] = v;
            sSgn[i]  = s;
        }
        __syncthreads();

        // ---------------- Dense env Jacobian columns -------------------------
        for (int m = lane; m < DOF; m += 32) {
            if (m < 3) {
                for (int kk = 0; kk < NK; ++kk) {
                    sJ[(3*kk+0)*JST + m] = sSgn[3*kk+0] * ((m == 0) ? 1.0f : 0.0f);
                    sJ[(3*kk+1)*JST + m] = sSgn[3*kk+1] * ((m == 1) ? 1.0f : 0.0f);
                    sJ[(3*kk+2)*JST + m] = sSgn[3*kk+2] * ((m == 2) ? 1.0f : 0.0f);
                }
            } else {
                int q = (m - 3) / 3;
                int i = (m - 3) - 3 * q;
                float r0 = sTheta[3+3*q], r1 = sTheta[4+3*q], r2 = sTheta[5+3*q];
                float t2  = r0*r0 + r1*r1 + r2*r2 + 1e-12f;
                float it2 = 1.0f / t2;
                float L00,L01,L02,L10,L11,L12,L20,L21,L22;
                rod9(r0, r1, r2, L00,L01,L02,L10,L11,L12,L20,L21,L22);
                // u = (I - R) e_i  (column i of I-R), ri = r[i]
                float u0, u1, u2, ri;
                if (i == 0)      { u0 = 1.0f - L00; u1 = -L10; u2 = -L20; ri = r0; }
                else if (i == 1) { u0 = -L01; u1 = 1.0f - L11; u2 = -L21; ri = r1; }
                else             { u0 = -L02; u1 = -L12; u2 = 1.0f - L22; ri = r2; }
                // w = (ri*r + r x u) / |r|^2  (local rotation axis)
                float w0 = (ri*r0 + (r1*u2 - r2*u1)) * it2;
                float w1 = (ri*r1 + (r2*u0 - r0*u2)) * it2;
                float w2 = (ri*r2 + (r0*u1 - r1*u0)) * it2;
                // world axis a = R_parent(q) * w
                float a0, a1, a2;
                if (q == 0) { a0 = w0; a1 = w1; a2 = w2; }
                else {
                    int pj = c_parents[q];
                    a0 = sRw[pj*9+0]*w0 + sRw[pj*9+1]*w1 + sRw[pj*9+2]*w2;
                    a1 = sRw[pj*9+3]*w0 + sRw[pj*9+4]*w1 + sRw[pj*9+5]*w2;
                    a2 = sRw[pj*9+6]*w0 + sRw[pj*9+7]*w1 + sRw[pj*9+8]*w2;
                }
                float pq0 = sPw[q*3+0], pq1 = sPw[q*3+1], pq2 = sPw[q*3+2];
                for (int kk = 0; kk < NK; ++kk) {
                    int k = c_lb[kk];
                    bool anc = false;
                    int t = k;
                    while (t >= 0) { if (t == q) { anc = true; break; } t = c_parents[t]; }
                    float cx = 0.0f, cy = 0.0f, cz = 0.0f;
                    if (anc) {
                        float d0 = sPw[k*3+0] - pq0;
                        float d1 = sPw[k*3+1] - pq1;
                        float d2 = sPw[k*3+2] - pq2;
                        cx = a1*d2 - a2*d1;
                        cy = a2*d0 - a0*d2;
                        cz = a0*d1 - a1*d0;
                    }
                    sJ[(3*kk+0)*JST + m] = sSgn[3*kk+0] * cx;
                    sJ[(3*kk+1)*JST + m] = sSgn[3*kk+1] * cy;
                    sJ[(3*kk+2)*JST + m] = sSgn[3*kk+2] * cz;
                }
            }
        }
        __syncthreads();

        // ---------------- rhs: b = -(J^T r_env + 0.01*(theta-q) on [3:]) ------
        for (int m = lane; m < MPAD; m += 32) {
            float g = 0.0f;
            if (m < DOF) {
                for (int row = 0; row < NR; ++row)
                    g += sJ[row*JST + m] * sRenv[row];
                if (m >= 3)
                    g += 0.01f * (sTheta[m] - qres[f * DOF + m]);
            }
            sB[m] = -g;
        }
        __syncthreads();

        // ---------------- A = J^T J via V_WMMA_F32_16X16X4_F32 ----------------
        {
            const int mn   = lane & 15;      // M / N within tile
            const int half = lane >> 4;      // lane half selects K pair / M offset
            for (int mt = 0; mt < 5; ++mt) {
                for (int nt = 0; nt < 5; ++nt) {
                    v8f acc = {};
                    const int ml = mt * 16 + mn;
                    const int nl = nt * 16 + mn;
                    #pragma unroll
                    for (int kt = 0; kt < 12; ++kt) {
                        const int kb = kt * 4 + 2 * half;   // K rows for this lane half
                        v2f av, bv;
                        av.x = sJ[(kb + 0) * JST + ml];     // A = J^T : (m,k) <- J[k][m]
                        av.y = sJ[(kb + 1) * JST + ml];
                        bv.x = sJ[(kb + 0) * JST + nl];     // B = J   : (k,n)
                        bv.y = sJ[(kb + 1) * JST + nl];
                        acc = __builtin_amdgcn_wmma_f32_16x16x4_f32(
                            false, av, false, bv, (short)0, acc, false, false);
                    }
                    #pragma unroll
                    for (int p = 0; p < 8; ++p) {
                        int M = mt * 16 + p + 8 * half;
                        int N = nt * 16 + mn;
                        sA[M * AST + N] = acc[p];
                    }
                }
            }
        }
        __syncthreads();

        // ---------------- Regularizers / padding identity ---------------------
        for (int i = lane; i < MPAD; i += 32) {
            if (i >= DOF) {
                sA[i * AST + i] = 1.0f;           // padding rows -> identity
            } else {
                float add = DAMPING;
                if (i >= 3) {
                    float s = sgnf(sTheta[i] - qres[f * DOF + i]);
                    add += 0.01f * s * s;          // (0.1*sign)^2 diag block
                }
                sA[i * AST + i] += add;
            }
        }
        __syncthreads();

        // ---------------- Gaussian elimination (SPD + damping: no pivoting) ---
        for (int k = 0; k < MPAD; ++k) {
            float pivinv = 1.0f / sA[k * AST + k];
            for (int i = k + 1 + lane; i < MPAD; i += 32) {
                float fac = sA[i * AST + k] * pivinv;
                sF[i] = fac;
                sB[i] -= fac * sB[k];
            }
            __syncthreads();
            const int n  = MPAD - k - 1;
            const int tot = n * n;
            for (int idx = lane; idx < tot; idx += 32) {
                int i = k + 1 + idx / n;
                int j = k + 1 + idx - (i - k - 1) * n;
                sA[i * AST + j] -= sF[i] * sA[k * AST + j];
            }
            __syncthreads();
        }
        // Back substitution
        for (int k = MPAD - 1; k >= 0; --k) {
            if (lane == 0) sDelta[k] = sB[k] / sA[k * AST + k];
            __syncthreads();
            for (int i = lane; i < k; i += 32)
                sB[i] -= sA[i * AST + k] * sDelta[k];
            __syncthreads();
        }

        // ---------------- Update ----------------------------------------------
        for (int m = lane; m < DOF; m += 32)
            sTheta[m] += STEP * sDelta[m];
        __syncthreads();
    }

    for (int m = lane; m < DOF; m += 32)
        out[f * DOF + m] = sTheta[m];
}

extern "C" void kernel_launch(void* const* d_in, const int* in_sizes, int n_in,
                              void* d_out, int out_size, void* d_ws, size_t ws_size,
                              hipStream_t stream)
{
    const float* theta0 = (const float*)d_in[0];   // theta_opt (NF,78)
    // d_in[1] = targeted_wposition, unused by the reference computation
    const float* tar    = (const float*)d_in[2];   // targeted_kposition (NF,15,3)
    const float* qres   = (const float*)d_in[3];   // q (NF,78)
    const float* offs   = (const float*)d_in[4];   // offsets (25,3)
    float* out          = (float*)d_out;           // (NF,78) f32

    const int nf = in_sizes[0] / DOF;
    lm_ik_kernel<<<nf, 32, 0, stream>>>(theta0, tar, qres, offs, out);
}